// ResBlock_48936857371129
// MI455X (gfx1250) — compile-verified
//
#include <hip/hip_runtime.h>
#include <math.h>

#define DFEAT 128

typedef float v2f __attribute__((ext_vector_type(2)));
typedef float v8f __attribute__((ext_vector_type(8)));

// Relaxed, agent-scope FP32 add -> should lower to global_atomic_add_f32 on gfx1250.
__device__ __forceinline__ void atomic_add_f32(float* p, float v) {
    __hip_atomic_fetch_add(p, v, __ATOMIC_RELAXED, __HIP_MEMORY_SCOPE_AGENT);
}

// ---------------- degree accumulation ----------------
__global__ __launch_bounds__(256) void deg_kernel(const int* __restrict__ src,
                                                  const int* __restrict__ dst,
                                                  float* __restrict__ outdeg,
                                                  float* __restrict__ indeg, int E) {
    int e = blockIdx.x * 256 + threadIdx.x;
    if (e < E) {
        atomic_add_f32(&outdeg[src[e]], 1.0f);
        atomic_add_f32(&indeg[dst[e]], 1.0f);
    }
}

__global__ __launch_bounds__(256) void norm_kernel(const float* __restrict__ outdeg,
                                                   const float* __restrict__ indeg,
                                                   float* __restrict__ srcn,
                                                   float* __restrict__ dstn, int n) {
    int i = blockIdx.x * 256 + threadIdx.x;
    if (i < n) {
        srcn[i] = rsqrtf(fmaxf(outdeg[i], 1.0f));
        dstn[i] = rsqrtf(fmaxf(indeg[i], 1.0f));
    }
}

// ---------------- edge scatter: agg[dst] += h[src] * srcn[src] ----------------
// One wave32 per edge; each lane handles a float4 (32 lanes * 4 = 128 features).
__global__ __launch_bounds__(256) void scatter_kernel(const float* __restrict__ h,
                                                      const float* __restrict__ srcn,
                                                      const int* __restrict__ src,
                                                      const int* __restrict__ dst,
                                                      float* __restrict__ agg, int E) {
    int wave = (blockIdx.x * 256 + threadIdx.x) >> 5;
    int lane = threadIdx.x & 31;
    if (wave >= E) return;
    int s = src[wave];
    int d = dst[wave];
    float ns = srcn[s];
    float4 v = ((const float4*)(h + (size_t)s * DFEAT))[lane];  // global_load_b128
    float* ap = agg + (size_t)d * DFEAT + lane * 4;
    atomic_add_f32(ap + 0, v.x * ns);
    atomic_add_f32(ap + 1, v.y * ns);
    atomic_add_f32(ap + 2, v.z * ns);
    atomic_add_f32(ap + 3, v.w * ns);
}

// ---------------- fused: out = SiLU(LN((agg * dstn) @ W + b)) [+ xskip] ----------------
// Block: 256 threads = 8 waves; block computes 16 rows x 128 cols.
// Wave w computes the 16x16 tile at columns [16w, 16w+16) with V_WMMA_F32_16X16X4_F32.
template <bool SKIP>
__global__ __launch_bounds__(256) void gemm_ln_silu_kernel(
    const float* __restrict__ agg, const float* __restrict__ dstn,
    const float* __restrict__ W, const float* __restrict__ bias,
    const float* __restrict__ gamma, const float* __restrict__ beta,
    const float* __restrict__ xskip, float* __restrict__ out, int nrows) {
    __shared__ float sA[16 * DFEAT];  // 8 KB: scaled A tile; reused as output tile
    __shared__ float sW[32 * DFEAT];  // 16 KB: one K-chunk of W

    const int t = threadIdx.x;
    const int rowBase = blockIdx.x * 16;

    // Stage A tile: 16 rows x 128 cols (scaled by dst_norm), 512 float4 slots.
    for (int j = 0; j < 2; ++j) {
        int slot = t + 256 * j;
        int r = slot >> 5;       // 32 float4 per row
        int c4 = slot & 31;
        int grow = rowBase + r;
        float4 v = make_float4(0.f, 0.f, 0.f, 0.f);
        if (grow < nrows) {
            v = ((const float4*)(agg + (size_t)grow * DFEAT))[c4];
            float nv = dstn[grow];
            v.x *= nv; v.y *= nv; v.z *= nv; v.w *= nv;
        }
        ((float4*)sA)[slot] = v;
    }

    const int lane = t & 31;
    const int wv = t >> 5;
    const int rl = lane & 15;    // A: row M;  B/C/D: column offset within tile
    const int g = lane >> 4;     // half-wave select
    const int nBase = wv * 16;

    // Accumulator pre-loaded with bias: every element in a lane has N = nBase + rl.
    float bcol = bias[nBase + rl];
    v8f c = {bcol, bcol, bcol, bcol, bcol, bcol, bcol, bcol};

    for (int kc = 0; kc < 4; ++kc) {
        __syncthreads();  // A tile ready (iter 0) / previous sW consumed
        // Stage W chunk rows [kc*32, kc*32+32), 1024 float4 slots.
        for (int j = 0; j < 4; ++j) {
            int slot = t + 256 * j;
            int kr = slot >> 5;
            int c4 = slot & 31;
            ((float4*)sW)[slot] = ((const float4*)(W + (size_t)(kc * 32 + kr) * DFEAT))[c4];
        }
        __syncthreads();
#pragma unroll
        for (int ks = 0; ks < 8; ++ks) {
            int kk = ks * 4;            // k within chunk
            int kg = kc * 32 + kk;      // global k (sA holds full K)
            v2f a, b;
            // A 16x4 fragment: lane rl = M; VGPR0/1 hold K = 2g / 2g+1 (ISA 7.12.2)
            a.x = sA[rl * DFEAT + kg + 2 * g];
            a.y = sA[rl * DFEAT + kg + 2 * g + 1];
            // B 4x16 fragment: rows striped across lanes; VGPR0/1 hold K rows 2g / 2g+1
            b.x = sW[(kk + 2 * g) * DFEAT + nBase + rl];
            b.y = sW[(kk + 2 * g + 1) * DFEAT + nBase + rl];
            c = __builtin_amdgcn_wmma_f32_16x16x4_f32(false, a, false, b,
                                                      (short)0, c, false, false);
        }
    }

    __syncthreads();
    // Spill accumulators into sA as the 16x128 output tile.
    // C/D layout: VGPR r -> M = r (lanes 0-15) / r+8 (lanes 16-31), N = rl.
#pragma unroll
    for (int r = 0; r < 8; ++r) {
        int m = r + 8 * g;
        sA[m * DFEAT + nBase + rl] = c[r];
    }
    __syncthreads();

    // LayerNorm + SiLU (+ skip). thread t: row = t>>4, owns 8 contiguous cols.
    const int row = t >> 4;
    const int seg = t & 15;
    const int base = seg * 8;
    float sum = 0.f, ssq = 0.f;
#pragma unroll
    for (int j = 0; j < 8; ++j) {
        float v = sA[row * DFEAT + base + j];
        sum += v;
        ssq += v * v;
    }
    // reduce across the 16 lanes of this row (wave32: two rows per wave, width 16)
    for (int off = 8; off > 0; off >>= 1) {
        sum += __shfl_xor(sum, off, 16);
        ssq += __shfl_xor(ssq, off, 16);
    }
    float mu = sum * (1.0f / DFEAT);
    float var = ssq * (1.0f / DFEAT) - mu * mu;
    float rstd = rsqrtf(var + 1e-5f);

    int grow = rowBase + row;
    if (grow < nrows) {
#pragma unroll
        for (int j = 0; j < 8; ++j) {
            int col = base + j;
            float v = sA[row * DFEAT + col];
            float y = (v - mu) * rstd * gamma[col] + beta[col];
            float sv = y / (1.0f + __expf(-y));  // SiLU
            if (SKIP) sv += xskip[(size_t)grow * DFEAT + col];
            out[(size_t)grow * DFEAT + col] = sv;
        }
    }
}

extern "C" void kernel_launch(void* const* d_in, const int* in_sizes, int n_in,
                              void* d_out, int out_size, void* d_ws, size_t ws_size,
                              hipStream_t stream) {
    const float* x   = (const float*)d_in[0];
    const float* W1  = (const float*)d_in[1];
    const float* b1  = (const float*)d_in[2];
    const float* g1  = (const float*)d_in[3];
    const float* be1 = (const float*)d_in[4];
    const float* W2  = (const float*)d_in[5];
    const float* b2  = (const float*)d_in[6];
    const float* g2  = (const float*)d_in[7];
    const float* be2 = (const float*)d_in[8];
    const int* src   = (const int*)d_in[9];
    const int* dst   = (const int*)d_in[10];

    const int N = in_sizes[0] / DFEAT;
    const int E = in_sizes[9];

    float* ws     = (float*)d_ws;
    float* srcn   = ws;                      // N
    float* dstn   = ws + (size_t)N;          // N
    float* outdeg = ws + 2 * (size_t)N;      // N
    float* indeg  = ws + 3 * (size_t)N;      // N
    float* agg    = ws + 4 * (size_t)N;      // N*D
    float* out    = (float*)d_out;           // final output; also holds h1 between layers

    // degrees -> norms
    hipMemsetAsync(outdeg, 0, 2 * (size_t)N * sizeof(float), stream);
    deg_kernel<<<(E + 255) / 256, 256, 0, stream>>>(src, dst, outdeg, indeg, E);
    norm_kernel<<<(N + 255) / 256, 256, 0, stream>>>(outdeg, indeg, srcn, dstn, N);

    // layer 1: aggregate x, then GEMM+LN+SiLU -> d_out (h1)
    hipMemsetAsync(agg, 0, (size_t)N * DFEAT * sizeof(float), stream);
    scatter_kernel<<<(E + 7) / 8, 256, 0, stream>>>(x, srcn, src, dst, agg, E);
    gemm_ln_silu_kernel<false><<<(N + 15) / 16, 256, 0, stream>>>(
        agg, dstn, W1, b1, g1, be1, nullptr, out, N);

    // layer 2: aggregate h1 (in d_out), then GEMM+LN+SiLU+residual -> d_out
    hipMemsetAsync(agg, 0, (size_t)N * DFEAT * sizeof(float), stream);
    scatter_kernel<<<(E + 7) / 8, 256, 0, stream>>>(out, srcn, src, dst, agg, E);
    gemm_ln_silu_kernel<true><<<(N + 15) / 16, 256, 0, stream>>>(
        agg, dstn, W2, b2, g2, be2, x, out, N);
}